// FocalLoss_44100724195748
// MI455X (gfx1250) — compile-verified
//
#include <hip/hip_runtime.h>
#include <cstdint>
#include <cstddef>

// Problem constants (match reference setup_inputs)
#define B_N 8
#define A_N 49104
#define C_N 90
#define M_N 32

typedef __attribute__((ext_vector_type(2))) float v2f;
typedef __attribute__((ext_vector_type(8))) float v8f;

// ---------------------------------------------------------------------------
// Single-op XOR-butterfly add using DS_SWIZZLE_B32 group-of-32 mode:
//   offset = xor_mask<<10 | or_mask<<5 | and_mask   (and=0x1f, or=0)
// One ds op per step vs. the ~6-op clamp+bpermute sequence __shfl_xor emits.
// ---------------------------------------------------------------------------
template <int PAT>
__device__ __forceinline__ float swz_add(float s) {
#if __has_builtin(__builtin_amdgcn_ds_swizzle)
    return s + __int_as_float(__builtin_amdgcn_ds_swizzle(__float_as_int(s), PAT));
#else
    return s + __shfl_xor(s, (PAT >> 10) & 0x1f, 32);
#endif
}

// ---------------------------------------------------------------------------
// Wave32 sum reduction.
// CDNA5 path: V_WMMA_F32_16X16X4_F32 with A = all-ones. Then
//   D[m,n] = sum_k B[k,n]   (column sums of B)
// Each lane's partial goes in B's first VGPR (second = 0), so the 32 partials
// occupy 32 distinct (k,n) slots of the 4x16 B matrix; column-sum totals are
// invariant to the exact lane->(k,n) mapping. After the WMMA every lane's
// d[0] holds colsum(lane%16); 4 xor-swizzle adds finish the reduction.
// Every lane returns the full 32-lane total.
// NOTE: WMMA requires EXEC == all ones -> only call from fully converged code.
// ---------------------------------------------------------------------------
__device__ __forceinline__ float wave_sum32(float x) {
#if defined(__gfx1250__) && __has_builtin(__builtin_amdgcn_wmma_f32_16x16x4_f32)
    v2f a; a[0] = 1.0f; a[1] = 1.0f;          // A = ones(16x4), both VGPRs
    v2f bm; bm[0] = x; bm[1] = 0.0f;          // 32 partials + 32 zeros
    v8f c = {};
    v8f d = __builtin_amdgcn_wmma_f32_16x16x4_f32(
        /*neg_a=*/false, a, /*neg_b=*/false, bm,
        /*c_mod=*/(short)0, c, /*reuse_a=*/false, /*reuse_b=*/false);
    float s = d[0];
    s = swz_add<0x041F>(s);   // SWAPX1
    s = swz_add<0x081F>(s);   // SWAPX2
    s = swz_add<0x101F>(s);   // SWAPX4
    s = swz_add<0x201F>(s);   // SWAPX8
    return s;
#else
    for (int m = 16; m >= 1; m >>= 1) x += __shfl_xor(x, m, 32);
    return x;
#endif
}

// ---------------------------------------------------------------------------
// Kernel 0: zero per-image accumulators (workspace is poisoned by harness).
// ---------------------------------------------------------------------------
__global__ void k_zero(float* clfS, float* regS, unsigned* nm) {
    int t = threadIdx.x;
    if (t < B_N) { clfS[t] = 0.0f; regS[t] = 0.0f; nm[t] = 0u; }
}

// ---------------------------------------------------------------------------
// Kernel 1: per-anchor IoU argmax over 32 boxes, match state, smooth-L1 reg.
// state[b*A+a] = gt class (0..89) if matched, -1 if unmatched, -2 if ignore.
// Accumulates per-image matched count and reg smooth-L1 sum.
// ---------------------------------------------------------------------------
__global__ void k_match(const float* __restrict__ anchors,
                        const float* __restrict__ ann,
                        const float* __restrict__ regs,
                        signed char* __restrict__ state,
                        float* __restrict__ regS,
                        unsigned* __restrict__ nm) {
    __shared__ float sAnn[M_N * 5];
    const int b   = blockIdx.y;
    const int tid = threadIdx.x;
    if (tid < M_N * 5) sAnn[tid] = ann[b * M_N * 5 + tid];
    __syncthreads();

    const int a = blockIdx.x * blockDim.x + tid;
    float regsum = 0.0f;
    bool matched = false;

    if (a < A_N) {
        // anchors are (y1, x1, y2, x2); boxes are (x1, y1, x2, y2, label)
        const float4 av = *(const float4*)(anchors + (size_t)a * 4);
        const float ay1 = av.x, ax1 = av.y, ay2 = av.z, ax2 = av.w;
        const float aarea = (ax2 - ax1) * (ay2 - ay1);

        float best = -2.0f;   // ious are >= 0 or exactly -1 (invalid box)
        int bestj = 0;
        #pragma unroll 4
        for (int j = 0; j < M_N; ++j) {
            const float bx1 = sAnn[j * 5 + 0];
            const float by1 = sAnn[j * 5 + 1];
            const float bx2 = sAnn[j * 5 + 2];
            const float by2 = sAnn[j * 5 + 3];
            const float lbl = sAnn[j * 5 + 4];
            float iw = fmaxf(fminf(ax2, bx2) - fmaxf(ax1, bx1), 0.0f);
            float ih = fmaxf(fminf(ay2, by2) - fmaxf(ay1, by1), 0.0f);
            float inter = iw * ih;
            float barea = (bx2 - bx1) * (by2 - by1);
            float uni = fmaxf(aarea + barea - inter, 1e-8f);
            float iou = inter / uni;
            if (lbl == -1.0f) iou = -1.0f;
            if (iou > best) { best = iou; bestj = j; }   // first-max tie break
        }

        matched = best > 0.5f;
        const bool unmatched = best < 0.4f;
        int st = -2;
        if (matched)        st = (int)sAnn[bestj * 5 + 4] - 1;  // 0..89
        else if (unmatched) st = -1;
        state[(size_t)b * A_N + a] = (signed char)st;

        if (matched) {
            const float aw  = ax2 - ax1, ah  = ay2 - ay1;
            const float acx = ax1 + 0.5f * aw, acy = ay1 + 0.5f * ah;
            const float bx1 = sAnn[bestj * 5 + 0];
            const float by1 = sAnn[bestj * 5 + 1];
            const float bx2 = sAnn[bestj * 5 + 2];
            const float by2 = sAnn[bestj * 5 + 3];
            const float gw_raw = bx2 - bx1, gh_raw = by2 - by1;
            const float gcx = bx1 + 0.5f * gw_raw;
            const float gcy = by1 + 0.5f * gh_raw;
            const float gw = fmaxf(gw_raw, 1.0f);
            const float gh = fmaxf(gh_raw, 1.0f);
            const float tv[4] = { (gcy - acy) / ah, (gcx - acx) / aw,
                                  __logf(gh / ah),  __logf(gw / aw) };
            const float4 r = *(const float4*)(regs + ((size_t)b * A_N + a) * 4);
            const float rv[4] = { r.x, r.y, r.z, r.w };
            #pragma unroll
            for (int k = 0; k < 4; ++k) {
                const float d = fabsf(9.0f * (rv[k] - tv[k]));
                regsum += (d < 1.0f) ? 0.5f * d * d : d - 0.5f;
            }
        }
    }

    // Fully converged here: safe to use WMMA reduction.
    const unsigned long long bm = __ballot(matched);
    const int mc = __popcll(bm);
    const float rs = wave_sum32(regsum);
    if ((tid & 31) == 0) {
        if (mc) atomicAdd(&nm[b], (unsigned)mc);
        atomicAdd(&regS[b], rs);
    }
}

// ---------------------------------------------------------------------------
// Kernel 2: focal classification loss (bandwidth-dominant: 141 MB stream).
// float4 grid-stride (A*C divisible by 4) -> global_load_b128, 512 B
// contiguous per wave per issue. Per-quad: one int-div for (anchor, class),
// rolled forward across at most one anchor boundary; state byte fetched
// 1-2x per quad. One atomic per wave per image.
// ---------------------------------------------------------------------------
__global__ void k_focal(const float* __restrict__ clf,
                        const signed char* __restrict__ state,
                        float* __restrict__ clfS) {
    const int b = blockIdx.y;
    const int elems  = A_N * C_N;       // 4,419,360
    const int elems4 = elems / 4;       // exactly divisible
    const float4* __restrict__ cp4 = (const float4*)(clf + (size_t)b * elems);
    const signed char* __restrict__ sp = state + (size_t)b * A_N;

    float partial = 0.0f;
    const int stride = gridDim.x * blockDim.x;
    for (int i4 = blockIdx.x * blockDim.x + threadIdx.x; i4 < elems4; i4 += stride) {
        if (i4 + stride < elems4) __builtin_prefetch(&cp4[i4 + stride], 0, 1);
        const float4 v = cp4[i4];
        const int base = i4 * 4;
        int a  = base / C_N;
        int c  = base - a * C_N;
        int st = (int)sp[a];
        const float pv[4] = { v.x, v.y, v.z, v.w };
        #pragma unroll
        for (int k = 0; k < 4; ++k) {
            if (c == C_N) { c = 0; ++a; st = (int)sp[a]; }
            float p = fminf(fmaxf(pv[k], 1e-4f), 1.0f - 1e-4f);
            float loss;
            if (st == -2) {                       // ignore band (0.4..0.5]
                loss = 0.0f;
            } else if (st == c) {                 // positive class of matched anchor
                const float om = 1.0f - p;
                loss = 0.25f * om * om * (-__logf(p));
            } else {                              // negative
                loss = 0.75f * p * p * (-__logf(1.0f - p));
            }
            partial += loss;
            ++c;
        }
    }

    const float s = wave_sum32(partial);      // converged: EXEC all ones
    if ((threadIdx.x & 31) == 0) atomicAdd(&clfS[b], s);
}

// ---------------------------------------------------------------------------
// Kernel 3: finalize the two scalars.
// ---------------------------------------------------------------------------
__global__ void k_final(const float* __restrict__ clfS,
                        const float* __restrict__ regS,
                        const unsigned* __restrict__ nm,
                        float* __restrict__ out) {
    const int t = threadIdx.x;
    float cl = 0.0f, rl = 0.0f;
    if (t < B_N) {
        const unsigned n = nm[t];
        const float dc = (float)(n > 1u ? n : 1u);           // max(1, nm)
        cl = clfS[t] / dc;
        rl = (n > 0u) ? regS[t] / (float)(4u * n) : 0.0f;    // max(1, 4*nm)
    }
    for (int m = 16; m >= 1; m >>= 1) {
        cl += __shfl_xor(cl, m, 32);
        rl += __shfl_xor(rl, m, 32);
    }
    if (t == 0) {
        out[0] = 1.0f  * (cl / (float)B_N);   // CLF_W * mean
        out[1] = 50.0f * (rl / (float)B_N);   // REG_W * mean
    }
}

// ---------------------------------------------------------------------------
extern "C" void kernel_launch(void* const* d_in, const int* in_sizes, int n_in,
                              void* d_out, int out_size, void* d_ws, size_t ws_size,
                              hipStream_t stream) {
    const float* clf     = (const float*)d_in[0];  // (B, A, C)
    const float* regs    = (const float*)d_in[1];  // (B, A, 4)
    const float* anchors = (const float*)d_in[2];  // (1, A, 4) -> flat A*4
    const float* ann     = (const float*)d_in[3];  // (B, M, 5)
    float* out = (float*)d_out;                    // 2 floats

    // Workspace layout: [B*A bytes: per-anchor state][B floats clf][B floats reg][B uints nm]
    signed char* state = (signed char*)d_ws;
    size_t stateBytes = (size_t)B_N * A_N;             // 392,832 B
    stateBytes = (stateBytes + 15) & ~(size_t)15;      // align accumulators
    float*    clfS = (float*)((char*)d_ws + stateBytes);
    float*    regS = clfS + B_N;
    unsigned* nm   = (unsigned*)(regS + B_N);

    k_zero<<<1, 32, 0, stream>>>(clfS, regS, nm);

    dim3 g1((A_N + 255) / 256, B_N);
    k_match<<<g1, 256, 0, stream>>>(anchors, ann, regs, state, regS, nm);

    dim3 g2(2048, B_N);   // ~2 float4 quads per thread over A*C per image
    k_focal<<<g2, 256, 0, stream>>>(clf, state, clfS);

    k_final<<<1, 32, 0, stream>>>(clfS, regS, nm, out);
}